// MGIoUPoly_63419487093095
// MI455X (gfx1250) — compile-verified
//
#include <hip/hip_runtime.h>
#include <math.h>

typedef float v2f __attribute__((ext_vector_type(2)));
typedef float v8f __attribute__((ext_vector_type(8)));

#define EPS        1e-9f
#define EDGE_EPS2  1e-12f   // (length > 1e-6)  <=>  (length^2 > 1e-12)

// compare-exchange on (ang, x, y) with compile-time indices -> stays in VGPRs
#define CSWAP(i, j)                                              \
    {                                                            \
        bool sw = ang[i] > ang[j];                               \
        float ta = ang[i], tb = ang[j];                          \
        ang[i] = sw ? tb : ta;  ang[j] = sw ? ta : tb;           \
        float xa = x[i], xb = x[j];                              \
        x[i] = sw ? xb : xa;    x[j] = sw ? xa : xb;             \
        float ya = y[i], yb = y[j];                              \
        y[i] = sw ? yb : ya;    y[j] = sw ? ya : yb;             \
    }

__device__ __forceinline__ void sort_by_angle(float* __restrict__ x,
                                              float* __restrict__ y) {
    float cx = 0.f, cy = 0.f;
#pragma unroll
    for (int k = 0; k < 8; ++k) { cx += x[k]; cy += y[k]; }
    cx *= 0.125f; cy *= 0.125f;
    float ang[8];
#pragma unroll
    for (int k = 0; k < 8; ++k) ang[k] = atan2f(y[k] - cy, x[k] - cx);
    // Batcher odd-even mergesort for 8 (19 comparators)
    CSWAP(0,1) CSWAP(2,3) CSWAP(4,5) CSWAP(6,7)
    CSWAP(0,2) CSWAP(1,3) CSWAP(4,6) CSWAP(5,7)
    CSWAP(1,2) CSWAP(5,6)
    CSWAP(0,4) CSWAP(1,5) CSWAP(2,6) CSWAP(3,7)
    CSWAP(2,4) CSWAP(3,5)
    CSWAP(1,2) CSWAP(3,4) CSWAP(5,6)
}

__global__ __launch_bounds__(256)
void mgiou_kernel(const float* __restrict__ pred,
                  const float* __restrict__ target,
                  const float* __restrict__ weight,
                  float* __restrict__ partial, int B)
{
    const int tid   = blockIdx.x * blockDim.x + threadIdx.x;
    const bool valid = tid < B;
    const int i     = valid ? tid : 0;          // clamp: keep all lanes active (WMMA needs EXEC==~0)

    // ---- load one polygon pair (coalesced float4) ----
    float px[8], py[8], tx[8], ty[8];
    const float4* p4 = (const float4*)(pred   + (size_t)i * 16);
    const float4* t4 = (const float4*)(target + (size_t)i * 16);
#pragma unroll
    for (int k = 0; k < 4; ++k) {
        float4 a = p4[k];
        px[2*k] = a.x; py[2*k] = a.y; px[2*k+1] = a.z; py[2*k+1] = a.w;
        float4 b = t4[k];
        tx[2*k] = b.x; ty[2*k] = b.y; tx[2*k+1] = b.z; ty[2*k+1] = b.w;
    }

    // ---- L1 fallback + all-zero-target detection ----
    float absT = 0.f, l1 = 0.f;
#pragma unroll
    for (int k = 0; k < 8; ++k) {
        absT += fabsf(tx[k]) + fabsf(ty[k]);
        l1   += fabsf(px[k] - tx[k]) + fabsf(py[k] - ty[k]);
    }
    const bool all_zero = (absT == 0.0f);

    // ---- sort vertices CCW by angle (only axes depend on the ordering;
    //      per-axis min/max over points is permutation-invariant) ----
    sort_by_angle(px, py);
    sort_by_angle(tx, ty);

    // ---- 16 axes: edge perpendiculars of sorted pred then sorted target ----
    float gsum = 0.f; int nvalid = 0;
#pragma unroll
    for (int k = 0; k < 16; ++k) {
        float ex, ey;
        if (k < 8) { int k2 = (k + 1) & 7; ex = px[k2] - px[k]; ey = py[k2] - py[k]; }
        else       { int kk = k - 8; int k2 = (kk + 1) & 7; ex = tx[k2] - tx[kk]; ey = ty[k2] - ty[kk]; }
        const float axx = -ey, axy = ex;
        const bool  mk  = (ex * ex + ey * ey) > EDGE_EPS2;

        float mn1 =  __builtin_huge_valf(), mx1 = -__builtin_huge_valf();
        float mn2 =  __builtin_huge_valf(), mx2 = -__builtin_huge_valf();
#pragma unroll
        for (int n = 0; n < 8; ++n) {
            float p1 = fmaf(px[n], axx, py[n] * axy);
            float p2 = fmaf(tx[n], axx, ty[n] * axy);
            mn1 = fminf(mn1, p1); mx1 = fmaxf(mx1, p1);
            mn2 = fminf(mn2, p2); mx2 = fmaxf(mx2, p2);
        }
        float inter     = fmaxf(fminf(mx1, mx2) - fmaxf(mn1, mn2), 0.f);
        float hull      = fmaxf(mx1, mx2) - fminf(mn1, mn2);
        float hull_safe = fmaxf(hull, EPS);
        float uni       = (mx1 - mn1) + (mx2 - mn2) - inter;
        float uni_safe  = fmaxf(uni, EPS);
        float g = inter / uni_safe - (hull_safe - uni_safe) / hull_safe;
        g = fminf(fmaxf(g, -1.f), 1.f);
        gsum   += mk ? g : 0.f;
        nvalid += mk ? 1 : 0;
    }
    const float nv    = (float)(nvalid > 0 ? nvalid : 1);
    const float mgiou = (1.0f - gsum / nv) * 0.5f;

    const float loss = all_zero ? l1 : mgiou;
    const float w    = weight[i];
    const float lw_v = valid ? loss * w : 0.f;
    const float w_v  = valid ? w        : 0.f;

    // ---- wave reduction of (Σ loss*w, Σ w) via one V_WMMA_F32_16X16X4_F32 ----
    // A (16x4): lanes 0-15 -> rows M=0..15 with K0=lw,K1=w ; lanes 16-31 -> K2=lw,K3=w.
    // B (4x16): columns 0-7 select (1,0,1,0) -> row-sums of lw ; columns 8-15 select (0,1,0,1) -> row-sums of w.
    // D[m,n<8] = lw[m]+lw[m+16] ; D[m,n>=8] = w[m]+w[m+16].
    const int lane = threadIdx.x & 31;
    v2f a; a[0] = lw_v; a[1] = w_v;
    const bool lowN = (lane & 15) < 8;
    v2f b; b[0] = lowN ? 1.f : 0.f; b[1] = lowN ? 0.f : 1.f;
    v8f c = {};
    v8f d = __builtin_amdgcn_wmma_f32_16x16x4_f32(
        /*neg_a=*/false, a, /*neg_b=*/false, b,
        /*c_mod=*/(short)0, c, /*reuse_a=*/false, /*reuse_b=*/false);

    float t = d[0] + d[1] + d[2] + d[3] + d[4] + d[5] + d[6] + d[7]; // sum rows in this half
    t += __shfl_xor(t, 16, 32);                                     // combine M halves
    // now lanes 0-7 (and 16-23) hold Σ lw over the wave; lanes 8-15 hold Σ w.

    __shared__ float redLW[8], redW[8];
    const int wv = threadIdx.x >> 5;
    if (lane == 0) redLW[wv] = t;
    if (lane == 8) redW[wv]  = t;
    __syncthreads();
    if (threadIdx.x == 0) {
        float slw = 0.f, sw = 0.f;
#pragma unroll
        for (int k = 0; k < 8; ++k) { slw += redLW[k]; sw += redW[k]; }
        partial[2 * blockIdx.x]     = slw;
        partial[2 * blockIdx.x + 1] = sw;
    }
}

__global__ __launch_bounds__(256)
void mgiou_finalize(const float* __restrict__ partial, int n, float* __restrict__ out)
{
    __shared__ float sLW[256], sW[256];
    float lw = 0.f, w = 0.f;
    for (int k = threadIdx.x; k < n; k += 256) {
        lw += partial[2 * k];
        w  += partial[2 * k + 1];
    }
    sLW[threadIdx.x] = lw; sW[threadIdx.x] = w;
    __syncthreads();
    for (int s = 128; s > 0; s >>= 1) {
        if ((int)threadIdx.x < s) {
            sLW[threadIdx.x] += sLW[threadIdx.x + s];
            sW[threadIdx.x]  += sW[threadIdx.x + s];
        }
        __syncthreads();
    }
    if (threadIdx.x == 0)
        out[0] = sLW[0] / fmaxf(sW[0], 1.0f);   // loss_weight == 1.0
}

extern "C" void kernel_launch(void* const* d_in, const int* in_sizes, int n_in,
                              void* d_out, int out_size, void* d_ws, size_t ws_size,
                              hipStream_t stream)
{
    const float* pred   = (const float*)d_in[0];
    const float* target = (const float*)d_in[1];
    const float* weight = (const float*)d_in[2];
    float*       out    = (float*)d_out;
    const int B = in_sizes[2];                  // weight is [B]

    const int blocks = (B + 255) / 256;
    float* partial = (float*)d_ws;              // 2 floats per block

    mgiou_kernel<<<blocks, 256, 0, stream>>>(pred, target, weight, partial, B);
    mgiou_finalize<<<1, 256, 0, stream>>>(partial, blocks, out);
}